// NLA_17626545782811
// MI455X (gfx1250) — compile-verified
//
#include <hip/hip_runtime.h>
#include <hip/hip_bf16.h>
#include <cstdint>

#ifndef __has_builtin
#define __has_builtin(x) 0
#endif

namespace {
constexpr int kB = 16384;
constexpr int kL = 50;
constexpr int kD = 128;
constexpr int kThreads = 256;   // 8 wave32
constexpr int kWaves = kThreads / 32;
constexpr int kRowsPerWaveMax = (kL + kWaves - 1) / kWaves;  // 7
}

// Native clang vector (HIP's float4 is a struct and is rejected by
// __builtin_nontemporal_store).
typedef float fv4 __attribute__((ext_vector_type(4)));

// Types matching the builtin prototype:
//   __builtin_amdgcn_global_load_async_to_lds_b128(v4i AS(1)*, v4i AS(3)*, Ii, Ii)
typedef int v4i_async __attribute__((vector_size(16)));
typedef __attribute__((address_space(1))) v4i_async* gbl_v4i_ptr;
typedef __attribute__((address_space(3))) v4i_async* lds_v4i_ptr;

// ---- CDNA5 async global->LDS copy (ASYNCcnt-tracked), 16B per lane --------
__device__ __forceinline__ void async_gather_b128(const float* gsrc, float* ldst) {
#if __has_builtin(__builtin_amdgcn_global_load_async_to_lds_b128)
    __builtin_amdgcn_global_load_async_to_lds_b128(
        (gbl_v4i_ptr)(uintptr_t)gsrc,
        (lds_v4i_ptr)(uint32_t)(uintptr_t)ldst,
        0 /*offset*/, 0 /*cpol*/);
#else
    uint32_t loff = (uint32_t)(uintptr_t)ldst;
    asm volatile("global_load_async_to_lds_b128 %0, %1, off"
                 :: "v"(loff), "v"(gsrc) : "memory");
#endif
}

__device__ __forceinline__ void wait_asynccnt0() {
#if __has_builtin(__builtin_amdgcn_s_wait_asynccnt)
    __builtin_amdgcn_s_wait_asynccnt(0);
#else
    asm volatile("s_wait_asynccnt 0" ::: "memory");
#endif
}

// One workgroup per batch element b.
__global__ __launch_bounds__(kThreads)
void NLA_17626545782811_kernel(const int* __restrict__ uid,
                               const int* __restrict__ rid,
                               const int* __restrict__ ing,
                               const float* __restrict__ user_table,
                               const float* __restrict__ recipe_table,
                               const float* __restrict__ ing_table,
                               const float* __restrict__ W_att,
                               float* __restrict__ out) {
    __shared__ float rows[kL * kD];   // 25.6 KB of staged ingredient rows
    __shared__ float logits[kL];
    __shared__ float weights[kL];

    const int b    = blockIdx.x;
    const int t    = threadIdx.x;
    const int wave = t >> 5;
    const int lane = t & 31;

    float* outb = out + (size_t)b * (3 * kD);

    // ---- Phase 1a: preload this wave's row indices in ONE vector load -----
    // Lane j holds the ingredient index for row l = wave + 8*j: one
    // global_load_b32 + one wait instead of 7 serialized index round-trips.
    int myidx = 0;
    {
        const int l = wave + kWaves * lane;         // lane == j
        if (lane < kRowsPerWaveMax && l < kL)
            myidx = ing[(size_t)b * kL + l];
    }

    // ---- Phase 1b: issue all async gathers back-to-back -------------------
    // readlane (not shfl/bpermute): constant source lane -> v_readlane_b32
    // into an SGPR, no DS-pipe traffic, scalar row index feeds the address.
    // One wave-wide b128 covers a full 512B ingredient row.
#pragma unroll
    for (int j = 0; j < kRowsPerWaveMax; ++j) {
        const int l = wave + kWaves * j;            // uniform within the wave
        if (l < kL) {
            const int idx = __builtin_amdgcn_readlane(myidx, j);
            const float* g = ing_table + (size_t)idx * kD + lane * 4;
            async_gather_b128(g, &rows[l * kD + lane * 4]);
        }
    }

    // ---- Phase 2 (overlapped): user/recipe rows pass straight through -----
    // Streaming output: non-temporal stores keep L2 for the embedding tables.
    if (wave == 0) {
        const fv4 v = *(const fv4*)(user_table + (size_t)uid[b] * kD + lane * 4);
        __builtin_nontemporal_store(v, (fv4*)(outb + lane * 4));
    } else if (wave == 1) {
        const fv4 v = *(const fv4*)(recipe_table + (size_t)rid[b] * kD + lane * 4);
        __builtin_nontemporal_store(v, (fv4*)(outb + kD + lane * 4));
    }

    wait_asynccnt0();        // each wave drains its own ASYNCcnt
    __syncthreads();         // LDS tile now complete for the whole block

    // ---- Phase 3: logits[l] = dot(rows[l], W_att) (bias is softmax-inv.) --
    const fv4 wv = *(const fv4*)(W_att + lane * 4);
#pragma unroll
    for (int j = 0; j < kRowsPerWaveMax; ++j) {
        const int l = wave + kWaves * j;
        if (l < kL) {
            const fv4 e = *(const fv4*)(&rows[l * kD + lane * 4]);
            float s = e.x * wv.x + e.y * wv.y + e.z * wv.z + e.w * wv.w;
            s += __shfl_xor(s, 16);
            s += __shfl_xor(s, 8);
            s += __shfl_xor(s, 4);
            s += __shfl_xor(s, 2);
            s += __shfl_xor(s, 1);
            if (lane == 0) logits[l] = s;
        }
    }
    __syncthreads();

    // ---- Phase 4: softmax over L=50 (trivial; single lane) ----------------
    if (t == 0) {
        float m = -INFINITY;
        for (int l = 0; l < kL; ++l) m = fmaxf(m, logits[l]);
        float ssum = 0.0f;
        for (int l = 0; l < kL; ++l) {
            const float e = __expf(logits[l] - m);
            weights[l] = e;
            ssum += e;
        }
        const float inv = 1.0f / ssum;
        for (int l = 0; l < kL; ++l) weights[l] *= inv;
    }
    __syncthreads();

    // ---- Phase 5: aggregated[d] = sum_l w[l] * rows[l][d] -----------------
    // 128 threads, stride-128 LDS rows -> conflict-free; weights[l] is an
    // LDS broadcast.
    if (t < kD) {
        float acc = 0.0f;
#pragma unroll 5
        for (int l = 0; l < kL; ++l) acc = fmaf(weights[l], rows[l * kD + t], acc);
        __builtin_nontemporal_store(acc, outb + 2 * kD + t);
    }
}

extern "C" void kernel_launch(void* const* d_in, const int* in_sizes, int n_in,
                              void* d_out, int out_size, void* d_ws, size_t ws_size,
                              hipStream_t stream) {
    (void)in_sizes; (void)n_in; (void)d_ws; (void)ws_size; (void)out_size;
    const int*   uid          = (const int*)d_in[0];
    const int*   rid          = (const int*)d_in[1];
    const int*   ing          = (const int*)d_in[2];
    const float* user_table   = (const float*)d_in[3];
    const float* recipe_table = (const float*)d_in[4];
    const float* ing_table    = (const float*)d_in[5];
    const float* W_att        = (const float*)d_in[6];
    // d_in[7] = b_att: constant bias is invariant under softmax -> unused.
    float* outp = (float*)d_out;

    NLA_17626545782811_kernel<<<dim3(kB), dim3(kThreads), 0, stream>>>(
        uid, rid, ing, user_table, recipe_table, ing_table, W_att, outp);
}